// TwoPlaneTensoRF_2164663517945
// MI455X (gfx1250) — compile-verified
//
#include <hip/hip_runtime.h>
#include <cstdint>
#include <cstddef>

namespace {
constexpr int kH = 512;
constexpr int kW = 512;
constexpr int kC = 128;          // N_COMP * OUT_CH
constexpr int kHW = kH * kW;
constexpr int kNPts = 524288;
constexpr int kWavesPerBlock = 4;
constexpr int kTotalWaves = 16384;
constexpr int kPtsPerWave = kNPts / kTotalWaves;  // 32

constexpr int kCT = 32;    // channels per transpose tile
constexpr int kHT = 128;   // hw positions per transpose tile
constexpr int kPad = 4;    // tile row stride 132 floats = 528 B (16B multiple)
}  // namespace

// ---------------------------------------------------------------------------
// Helpers
// ---------------------------------------------------------------------------
__device__ __forceinline__ float4 lerp4(float4 a, float4 b, float w) {
  return make_float4(fmaf(b.x - a.x, w, a.x), fmaf(b.y - a.y, w, a.y),
                     fmaf(b.z - a.z, w, a.z), fmaf(b.w - a.w, w, a.w));
}

// t* = (coord - lo) / (hi - lo); replicate reference arithmetic exactly.
__device__ __forceinline__ void calc_corner(float tx, float ty, int& off00,
                                            float& wx, float& wy) {
  const float gx = tx * 2.0f - 1.0f;
  const float gy = ty * 2.0f - 1.0f;
  const float ix = (gx + 1.0f) * 0.5f * (float)(kW - 1);
  const float iy = (gy + 1.0f) * 0.5f * (float)(kH - 1);
  int x0 = (int)floorf(ix);
  x0 = x0 < 0 ? 0 : (x0 > kW - 2 ? kW - 2 : x0);
  int y0 = (int)floorf(iy);
  y0 = y0 < 0 ? 0 : (y0 > kH - 2 ? kH - 2 : y0);
  wx = ix - (float)x0;
  wy = iy - (float)y0;
  off00 = y0 * kW + x0;
}

// One async 512B corner copy: 32 lanes x b128, global -> LDS, ASYNCcnt-tracked.
__device__ __forceinline__ void async_copy512(const float* gbase,
                                              unsigned lds_byte, int lane) {
  const unsigned long long ga =
      (unsigned long long)(uintptr_t)(gbase + lane * 4);
  const unsigned la = lds_byte + (unsigned)lane * 16u;
  asm volatile("global_load_async_to_lds_b128 %0, %1, off" ::"v"(la), "v"(ga)
               : "memory");
}

// ---------------------------------------------------------------------------
// Kernel 1: [C][HW] -> [HW][C] transpose.
// 32c x 128hw tile; b128 global reads, b128 LDS stores (16B-aligned padded
// rows), b128 global writes with lanes varying channel fastest so each 8-lane
// group emits one fully-utilized 128B line.
// ---------------------------------------------------------------------------
__global__ __launch_bounds__(256) void transpose_kernel(
    const float* __restrict__ src, float* __restrict__ dst) {
  __shared__ float tile[kCT][kHT + kPad];
  const int t = (int)threadIdx.x;
  const int hw_base = (int)blockIdx.x * kHT;
  const int c_base = (int)blockIdx.y * kCT;

  // Read phase: lane covers hw = lane*4, c0 = t>>5 (+8 per pass).
  {
    const int lane = t & 31;
    const int c0 = t >> 5;  // 0..7
#pragma unroll
    for (int k = 0; k < 4; ++k) {
      const int c = c0 + k * 8;
      const float4 v = *(const float4*)(src + (size_t)(c_base + c) * kHW +
                                        hw_base + lane * 4);
      *(float4*)&tile[c][lane * 4] = v;  // row stride 528B -> 16B aligned
    }
  }
  __syncthreads();

  // Write phase: c4 = (t&7)*4 fastest across lanes, hw = t>>3 (+32 per pass).
  {
    const int c4 = (t & 7) * 4;
    const int hw0 = t >> 3;  // 0..31
#pragma unroll
    for (int k = 0; k < 4; ++k) {
      const int hw = hw0 + k * 32;
      float4 v;
      v.x = tile[c4 + 0][hw];
      v.y = tile[c4 + 1][hw];
      v.z = tile[c4 + 2][hw];
      v.w = tile[c4 + 3][hw];
      *(float4*)(dst + (size_t)(hw_base + hw) * kC + c_base + c4) = v;
    }
  }
}

// ---------------------------------------------------------------------------
// Kernel 2: gather + bilinear + product + 16-comp reduce + sigmoid.
// One point per wave per iteration; 8 async-to-LDS corner copies per point,
// double-buffered; next point's corner bases/weights carried in registers.
// ---------------------------------------------------------------------------
__global__ __launch_bounds__(kWavesPerBlock * 32) void gather_kernel(
    const float* __restrict__ x, const float* __restrict__ uvT,
    const float* __restrict__ stT, const float* __restrict__ bnd,
    float* __restrict__ out) {
  __shared__ float lds[kWavesPerBlock][2][8 * kC];  // 32 KB static

  const int lane = (int)(threadIdx.x & 31u);
  const int wv = (int)(threadIdx.x >> 5u);
  const int gw = (int)blockIdx.x * kWavesPerBlock + wv;
  const int p0 = gw * kPtsPerWave;

  const float lo0 = bnd[0], lo1 = bnd[1], lo2 = bnd[2], lo3 = bnd[3];
  const float r0 = 1.0f / (bnd[4] - lo0);
  const float r1 = 1.0f / (bnd[5] - lo1);
  const float r2 = 1.0f / (bnd[6] - lo2);
  const float r3 = 1.0f / (bnd[7] - lo3);

  const unsigned lds_base = (unsigned)(uintptr_t)(&lds[wv][0][0]);

  auto corners = [&](int p, int& ubase, float& uwx, float& uwy, int& sbase,
                     float& swx, float& swy) {
    const float4 xc = *(const float4*)(x + (size_t)p * 4);
    int uoff, soff;
    calc_corner((xc.x - lo0) * r0, (xc.y - lo1) * r1, uoff, uwx, uwy);
    calc_corner((xc.z - lo2) * r2, (xc.w - lo3) * r3, soff, swx, swy);
    ubase = uoff * kC;
    sbase = soff * kC;
  };

  auto stage = [&](int ub, int sb, int buf) {
    const unsigned lb = lds_base + (unsigned)buf * (8u * kC * 4u);
    async_copy512(uvT + ub, lb + 0u * 512u, lane);
    async_copy512(uvT + ub + kC, lb + 1u * 512u, lane);
    async_copy512(uvT + ub + kW * kC, lb + 2u * 512u, lane);
    async_copy512(uvT + ub + kW * kC + kC, lb + 3u * 512u, lane);
    async_copy512(stT + sb, lb + 4u * 512u, lane);
    async_copy512(stT + sb + kC, lb + 5u * 512u, lane);
    async_copy512(stT + sb + kW * kC, lb + 6u * 512u, lane);
    async_copy512(stT + sb + kW * kC + kC, lb + 7u * 512u, lane);
  };

  // Current point's corner state, carried in registers.
  int cu_ub = 0, cu_sb = 0;
  float cu_uwx = 0.f, cu_uwy = 0.f, cu_swx = 0.f, cu_swy = 0.f;
  if (p0 < kNPts) {
    corners(p0, cu_ub, cu_uwx, cu_uwy, cu_sb, cu_swx, cu_swy);
    stage(cu_ub, cu_sb, 0);
  }

  for (int i = 0; i < kPtsPerWave; ++i) {
    const int p = p0 + i;
    if (p >= kNPts) break;

    int nx_ub = 0, nx_sb = 0;
    float nx_uwx = 0.f, nx_uwy = 0.f, nx_swx = 0.f, nx_swy = 0.f;
    const bool more = (i + 1 < kPtsPerWave) && (p + 1 < kNPts);
    if (more) {
      corners(p + 1, nx_ub, nx_uwx, nx_uwy, nx_sb, nx_swx, nx_swy);
      stage(nx_ub, nx_sb, (i + 1) & 1);
      // 16 in flight; async loads complete in order -> oldest 8 (this buf) done
      asm volatile("s_wait_asynccnt 0x8" ::: "memory");
    } else {
      asm volatile("s_wait_asynccnt 0x0" ::: "memory");
    }

    const float* buf = &lds[wv][i & 1][0];
    const int lo = lane * 4;  // lane l holds channels 4l..4l+3
    const float4 u00 = *(const float4*)(buf + 0 * kC + lo);
    const float4 u01 = *(const float4*)(buf + 1 * kC + lo);
    const float4 u10 = *(const float4*)(buf + 2 * kC + lo);
    const float4 u11 = *(const float4*)(buf + 3 * kC + lo);
    const float4 s00 = *(const float4*)(buf + 4 * kC + lo);
    const float4 s01 = *(const float4*)(buf + 5 * kC + lo);
    const float4 s10 = *(const float4*)(buf + 6 * kC + lo);
    const float4 s11 = *(const float4*)(buf + 7 * kC + lo);

    const float4 u =
        lerp4(lerp4(u00, u01, cu_uwx), lerp4(u10, u11, cu_uwx), cu_uwy);
    const float4 s =
        lerp4(lerp4(s00, s01, cu_swx), lerp4(s10, s11, cu_swx), cu_swy);
    float4 v = make_float4(u.x * s.x, u.y * s.y, u.z * s.z, u.w * s.w);

    // Parity-preserving butterfly: even lanes sum channels c%8 in 0..3,
    // odd lanes c%8 in 4..7. Lane0 -> out[0..3], lane1 -> out[4..7].
#pragma unroll
    for (int m = 2; m <= 16; m <<= 1) {
      v.x += __shfl_xor(v.x, m, 32);
      v.y += __shfl_xor(v.y, m, 32);
      v.z += __shfl_xor(v.z, m, 32);
      v.w += __shfl_xor(v.w, m, 32);
    }

    if (lane < 2) {
      float4 r;
      r.x = 1.0f / (1.0f + __expf(-v.x));
      r.y = 1.0f / (1.0f + __expf(-v.y));
      r.z = 1.0f / (1.0f + __expf(-v.z));
      r.w = 1.0f / (1.0f + __expf(-v.w));
      *(float4*)(out + (size_t)p * 8 + lane * 4) = r;
    }

    cu_ub = nx_ub;
    cu_sb = nx_sb;
    cu_uwx = nx_uwx;
    cu_uwy = nx_uwy;
    cu_swx = nx_swx;
    cu_swy = nx_swy;
  }
}

// ---------------------------------------------------------------------------
// Fallback (scratch too small): direct strided gather from [C,H,W] layout.
// ---------------------------------------------------------------------------
__global__ __launch_bounds__(256) void fallback_kernel(
    const float* __restrict__ x, const float* __restrict__ uv,
    const float* __restrict__ st, const float* __restrict__ bnd,
    float* __restrict__ out) {
  const int p = (int)(blockIdx.x * blockDim.x + threadIdx.x);
  if (p >= kNPts) return;
  const float4 xc = *(const float4*)(x + (size_t)p * 4);
  const float t0 = (xc.x - bnd[0]) / (bnd[4] - bnd[0]);
  const float t1 = (xc.y - bnd[1]) / (bnd[5] - bnd[1]);
  const float t2 = (xc.z - bnd[2]) / (bnd[6] - bnd[2]);
  const float t3 = (xc.w - bnd[3]) / (bnd[7] - bnd[3]);
  int uoff, soff;
  float uwx, uwy, swx, swy;
  calc_corner(t0, t1, uoff, uwx, uwy);
  calc_corner(t2, t3, soff, swx, swy);

  float acc[8];
#pragma unroll
  for (int o = 0; o < 8; ++o) acc[o] = 0.0f;

  for (int c = 0; c < kC; ++c) {
    const float* pu = uv + (size_t)c * kHW + uoff;
    const float ut = fmaf(pu[1] - pu[0], uwx, pu[0]);
    const float ubt = fmaf(pu[kW + 1] - pu[kW], uwx, pu[kW]);
    const float uval = fmaf(ubt - ut, uwy, ut);
    const float* ps = st + (size_t)c * kHW + soff;
    const float st_ = fmaf(ps[1] - ps[0], swx, ps[0]);
    const float sbt = fmaf(ps[kW + 1] - ps[kW], swx, ps[kW]);
    const float sval = fmaf(sbt - st_, swy, st_);
    acc[c & 7] += uval * sval;
  }
#pragma unroll
  for (int o = 0; o < 8; ++o)
    out[(size_t)p * 8 + o] = 1.0f / (1.0f + __expf(-acc[o]));
}

// ---------------------------------------------------------------------------
// Launch
// ---------------------------------------------------------------------------
extern "C" void kernel_launch(void* const* d_in, const int* in_sizes, int n_in,
                              void* d_out, int out_size, void* d_ws,
                              size_t ws_size, hipStream_t stream) {
  (void)in_sizes;
  (void)n_in;
  (void)out_size;
  const float* x = (const float*)d_in[0];
  const float* uv = (const float*)d_in[1];
  const float* st = (const float*)d_in[2];
  const float* bnd = (const float*)d_in[3];
  float* out = (float*)d_out;

  const size_t plane_bytes = (size_t)kC * kHW * sizeof(float);  // 128 MB
  if (ws_size >= 2 * plane_bytes) {
    float* uvT = (float*)d_ws;
    float* stT = (float*)((char*)d_ws + plane_bytes);
    dim3 tb(256, 1, 1);
    dim3 tg(kHW / kHT, kC / kCT, 1);
    transpose_kernel<<<tg, tb, 0, stream>>>(uv, uvT);
    transpose_kernel<<<tg, tb, 0, stream>>>(st, stT);
    gather_kernel<<<kTotalWaves / kWavesPerBlock, kWavesPerBlock * 32, 0,
                    stream>>>(x, uvT, stT, bnd, out);
  } else {
    fallback_kernel<<<(kNPts + 255) / 256, 256, 0, stream>>>(x, uv, st, bnd,
                                                             out);
  }
}